// SquaredMonarchMatrix_76965813944739
// MI455X (gfx1250) — compile-verified
//
#include <hip/hip_runtime.h>

typedef __attribute__((ext_vector_type(2))) float v2f;
typedef __attribute__((ext_vector_type(8))) float v8f;

#define TT 16                 // batch elements per workgroup (WMMA N dim)
#define MSTRIDE 4100          // floats per batch elem in main LDS (64*64 + 4 pad, 16B aligned)
#define OSTRIDE 516           // floats per batch elem in out-chunk LDS (64*8 + 4 pad)

// Low 32 bits of a flat pointer into LDS == LDS byte address (shared aperture
// keeps the segment offset in addr[31:0] per CDNA5 ISA 10.2).
__device__ __forceinline__ unsigned lds_addr(const void* p) {
    return (unsigned)(reinterpret_cast<uintptr_t>(p));
}

// D = A(16x4) x B(4x16) + C, f32. Layouts per CDNA5 ISA 7.12.2:
//  A: lane m = lane&15, VGPR v holds K = v + 2*(lane>>4)
//  B: lane n = lane&15, VGPR v holds K = v + 2*(lane>>4)
//  C/D: lane n = lane&15, VGPR v holds M = v + 8*(lane>>4)
template <int STAGE>
__global__ __launch_bounds__(256) void monarch_stage_kernel(
    const float* __restrict__ in,   // [batch][4096]
    const float* __restrict__ prm,  // P[STAGE]: [64][64][64]
    float* __restrict__ out)        // [batch][4096]
{
    __shared__ float ldsIn[TT * MSTRIDE];
    __shared__ float ldsOut[TT * OSTRIDE];

    const int tid  = threadIdx.x;
    const int lane = tid & 31;
    const int wave = tid >> 5;      // 0..7
    const int r    = lane & 15;     // N-lane / M-lane / batch-in-tile
    const int h    = lane >> 4;     // K-half selector
    const long t0  = (long)blockIdx.x * TT;

    const float* gInTile  = in + t0 * 4096;   // SGPR base for async loads
    float*       gOutTile = out + t0 * 4096;  // SGPR base for async stores

    // ---- async, fully-coalesced contiguous tile load: 16 x 4096 floats ----
    for (int it = tid; it < TT * 1024; it += 256) {
        const int t  = it >> 10;
        const int c4 = it & 1023;
        const unsigned goff  = (unsigned)((t * 4096 + c4 * 4) * 4);  // bytes, < 256KB
        const unsigned laddr = lds_addr(&ldsIn[t * MSTRIDE + c4 * 4]);
        asm volatile("global_load_async_to_lds_b128 %0, %1, %2"
                     :
                     : "v"(laddr), "v"(goff), "s"(gInTile)
                     : "memory");
    }
    asm volatile("s_wait_asynccnt 0x0" ::: "memory");
    __syncthreads();

    for (int rr = 0; rr < 8; ++rr) {
        const int n = rr * 8 + wave;               // this wave's block column
        const float* pA = prm + (long)n * 4096;    // P[n] : 64x64 row-major [i][j]
        const float* lB = &ldsIn[r * MSTRIDE];     // batch element r of this tile

        v8f acc0 = {}, acc1 = {}, acc2 = {}, acc3 = {};

#pragma unroll
        for (int j0 = 0; j0 < 64; j0 += 4) {
            // B fragment: B[k, t=r] = input element for batch r at K = j0 + 2h + {0,1}
            v2f b;
            if (STAGE == 0) {
                // stage 1 reads column n: flat (j*64 + n) -> ds_load_2addr_stride64
                b.x = lB[(j0 + 2 * h) * 64 + n];
                b.y = lB[(j0 + 2 * h + 1) * 64 + n];
            } else {
                // stage 2 reads row n: flat (n*64 + j) (contiguous -> ds_load_b64)
                const float2 bv =
                    *reinterpret_cast<const float2*>(&lB[n * 64 + j0 + 2 * h]);
                b.x = bv.x;
                b.y = bv.y;
            }
            // A fragments for the 4 i-tiles: A[m=r, k] = P[n, i0+r, j0+2h+{0,1}]
            const float* pArow = pA + (long)r * 64 + j0 + 2 * h;
            v2f a0 = *reinterpret_cast<const v2f*>(pArow + 0);     // i0 = 0
            v2f a1 = *reinterpret_cast<const v2f*>(pArow + 1024);  // i0 = 16
            v2f a2 = *reinterpret_cast<const v2f*>(pArow + 2048);  // i0 = 32
            v2f a3 = *reinterpret_cast<const v2f*>(pArow + 3072);  // i0 = 48

            acc0 = __builtin_amdgcn_wmma_f32_16x16x4_f32(false, a0, false, b,
                                                         (short)0, acc0, false, false);
            acc1 = __builtin_amdgcn_wmma_f32_16x16x4_f32(false, a1, false, b,
                                                         (short)0, acc1, false, false);
            acc2 = __builtin_amdgcn_wmma_f32_16x16x4_f32(false, a2, false, b,
                                                         (short)0, acc2, false, false);
            acc3 = __builtin_amdgcn_wmma_f32_16x16x4_f32(false, a3, false, b,
                                                         (short)0, acc3, false, false);
        }

        // WAR: previous round's async stores must be done reading ldsOut
        asm volatile("s_wait_asynccnt 0x0" ::: "memory");
        __syncthreads();

        // Stage D frags: element (t=r, i = i0 + v + 8h, nc = wave) -> ldsOut[t][i][nc]
        {
            float* so = &ldsOut[r * OSTRIDE + wave];
#pragma unroll
            for (int v = 0; v < 8; ++v) {
                const int ib = v + 8 * h;
                so[(0 + ib) * 8]  = acc0[v];
                so[(16 + ib) * 8] = acc1[v];
                so[(32 + ib) * 8] = acc2[v];
                so[(48 + ib) * 8] = acc3[v];
            }
        }
        __syncthreads();  // staging visible to all waves

        // Async chunk writeback: out[t][i*64 + rr*8 + {0..7}], 16B per op
        for (int p = tid; p < TT * 128; p += 256) {
            const int t    = p >> 7;
            const int q    = p & 127;   // 16B-chunk index within batch elem
            const int i    = q >> 1;
            const int half = q & 1;
            const unsigned laddr =
                lds_addr(&ldsOut[t * OSTRIDE + i * 8 + half * 4]);
            const unsigned goff =
                (unsigned)((t * 4096 + i * 64 + rr * 8 + half * 4) * 4);
            asm volatile("global_store_async_from_lds_b128 %0, %1, %2"
                         :
                         : "v"(goff), "v"(laddr), "s"(gOutTile)
                         : "memory");
        }
        // Final round's stores are covered by S_ENDPGM's implicit wait-idle;
        // intermediate rounds are covered by the s_wait_asynccnt above.
    }
}

extern "C" void kernel_launch(void* const* d_in, const int* in_sizes, int n_in,
                              void* d_out, int out_size, void* d_ws, size_t ws_size,
                              hipStream_t stream) {
    const float* x   = (const float*)d_in[0];   // [8,2048,4096] f32
    const float* prm = (const float*)d_in[1];   // [2,64,64,64]  f32
    float* out = (float*)d_out;

    const int batch = in_sizes[0] / 4096;       // 16384
    dim3 grid(batch / TT), block(256);

    // Stage 1: x -> H (stored into d_out as intermediate)
    monarch_stage_kernel<0><<<grid, block, 0, stream>>>(x, prm, out);
    // Stage 2: H -> Y, in-place on d_out (each WG snapshots its whole tile to LDS first)
    monarch_stage_kernel<1><<<grid, block, 0, stream>>>(out, prm + 64 * 64 * 64, out);
}